// MalGAT_52836687675576
// MI455X (gfx1250) — compile-verified
//
#include <hip/hip_runtime.h>
#include <hip/hip_bf16.h>
#include <math.h>

// Problem constants (match reference)
#define KK 2
#define BB 4
#define VV 2000
#define DD 64
#define HH 4
#define DHH 64
#define PP 128
#define ALPHA 0.2f
#define FNEG (-3.0e38f)
#define MAXN 2048   // n_present <= V = 2000

typedef __attribute__((ext_vector_type(2))) float v2f;
typedef __attribute__((ext_vector_type(8))) float v8f;

static __device__ __forceinline__ float lrelu(float v) {
    return v >= 0.0f ? v : ALPHA * v;
}

static __device__ __forceinline__ v8f wmma_f32_16x16x4(v2f a, v2f b, v8f c) {
    // 8-arg pattern: (neg_a, A, neg_b, B, c_mod, C, reuse_a, reuse_b)
    return __builtin_amdgcn_wmma_f32_16x16x4_f32(false, a, false, b, (short)0, c,
                                                 false, false);
}

// ---------------------------------------------------------------------------
// Kernel 1: EW[h][v][o] = sum_f E[v][f] * W[h][f][o]   (batch-independent)
// grid (V/16, H), block 128 (4 waves); wave w -> o-tile w. fp32 WMMA 16x16x4.
// ---------------------------------------------------------------------------
__global__ void malgat_ew_kernel(const float* __restrict__ E,
                                 const float* __restrict__ W,
                                 float* __restrict__ EW) {
    const int h    = blockIdx.y;
    const int v0   = blockIdx.x * 16;
    const int lane = threadIdx.x & 31;
    const int wave = threadIdx.x >> 5;
    const int o0   = wave * 16;
    const int rc   = lane & 15;               // A row / B col within tile
    const int koff = (lane < 16) ? 0 : 2;     // ISA A/B f32 layout half-wave K split

    v8f acc = {};
    for (int k = 0; k < DD; k += 4) {
        v2f a, b;
        const float* Erow = E + (size_t)(v0 + rc) * DD + k + koff;
        a.x = Erow[0];
        a.y = Erow[1];
        const float* Wrow = W + ((size_t)h * DD + k + koff) * DHH + o0 + rc;
        b.x = Wrow[0];
        b.y = Wrow[DHH];
        acc = wmma_f32_16x16x4(a, b, acc);
    }
    for (int j = 0; j < 8; ++j) {
        int row = j + ((lane < 16) ? 0 : 8);
        EW[((size_t)h * VV + v0 + row) * DHH + o0 + rc] = acc[j];
    }
}

// ---------------------------------------------------------------------------
// Kernel 2: f1[h][v] = EW[h][v]·a1[h],  f2 likewise (batch-independent)
// ---------------------------------------------------------------------------
__global__ void malgat_f12_kernel(const float* __restrict__ EW,
                                  const float* __restrict__ a1,
                                  const float* __restrict__ a2,
                                  float* __restrict__ f1,
                                  float* __restrict__ f2) {
    int t = blockIdx.x * blockDim.x + threadIdx.x;
    if (t >= HH * VV) return;
    int h = t / VV;
    const float* row = EW + (size_t)t * DHH;
    const float* p1 = a1 + h * DHH;
    const float* p2 = a2 + h * DHH;
    float s1 = 0.0f, s2 = 0.0f;
    for (int o = 0; o < DHH; ++o) {
        float e = row[o];
        s1 += e * p1[o];
        s2 += e * p2[o];
    }
    f1[t] = s1;
    f2[t] = s2;
}

// ---------------------------------------------------------------------------
// Kernel 3: deterministic presence compaction per (k,b) + latent-max init.
// latent init = 0 if any node absent (absent rows contribute 0 to the max),
// else -inf. Serial scan by thread 0 => fixed deterministic index order.
// ---------------------------------------------------------------------------
__global__ void malgat_compact_kernel(const float* __restrict__ x,
                                      int* __restrict__ idx,
                                      int* __restrict__ counts,
                                      float* __restrict__ latent) {
    int kb = blockIdx.x;  // kb = k*B + b
    __shared__ int sc;
    if (threadIdx.x == 0) {
        const float* xb = x + (size_t)kb * VV;
        int* list = idx + (size_t)kb * VV;
        int c = 0;
        for (int v = 0; v < VV; ++v)
            if (xb[v] > 0.0f) list[c++] = v;
        counts[kb] = c;
        sc = c;
    }
    __syncthreads();
    float init = (sc < VV) ? 0.0f : FNEG;
    for (int f = threadIdx.x; f < HH * DHH; f += blockDim.x)
        latent[(size_t)kb * (HH * DHH) + f] = init;
}

// ---------------------------------------------------------------------------
// Kernel 4: compacted masked GAT attention per (k,b,h).
// Pass 1: per-row softmax stats (max, 1/denom) over present u.
// Pass 2: out[j][o] = sum_u P[j][u]*EW[u][o] via fp32 WMMA 16x16x4,
//         A = probability tile built on the fly, B = gathered EW rows.
// elu + atomicMax (order-independent => deterministic) into latent.
// ---------------------------------------------------------------------------
__global__ void malgat_attn_kernel(const float* __restrict__ EW,
                                   const float* __restrict__ f1,
                                   const float* __restrict__ f2,
                                   const int* __restrict__ idx,
                                   const int* __restrict__ counts,
                                   float* __restrict__ latent) {
    __shared__ int   sidx[MAXN];
    __shared__ float f1p[MAXN];
    __shared__ float f2p[MAXN];
    __shared__ float mrow[MAXN];
    __shared__ float rden[MAXN];

    const int kb = blockIdx.x;
    const int h  = blockIdx.y;
    const int n  = counts[kb];
    if (n == 0) return;

    const int tid  = threadIdx.x;
    const int lane = tid & 31;
    const int wave = tid >> 5;
    const int nw   = blockDim.x >> 5;

    const int* list = idx + (size_t)kb * VV;
    for (int i = tid; i < n; i += blockDim.x) {
        int v = list[i];
        sidx[i] = v;
        f1p[i] = f1[h * VV + v];
        f2p[i] = f2[h * VV + v];
    }
    __syncthreads();

    // Pass 1: softmax stats per present row j (wave per row, lanes over u)
    for (int j = wave; j < n; j += nw) {
        float fj = f1p[j];
        float mx = FNEG;
        for (int u = lane; u < n; u += 32)
            mx = fmaxf(mx, lrelu(fj + f2p[u]));
        for (int off = 16; off > 0; off >>= 1)
            mx = fmaxf(mx, __shfl_xor(mx, off, 32));
        float sum = 0.0f;
        for (int u = lane; u < n; u += 32)
            sum += expf(lrelu(fj + f2p[u]) - mx);
        for (int off = 16; off > 0; off >>= 1)
            sum += __shfl_xor(sum, off, 32);
        if (lane == 0) {
            mrow[j] = mx;
            rden[j] = 1.0f / sum;
        }
    }
    __syncthreads();

    // Pass 2: WMMA attn @ EW_present
    const int rc   = lane & 15;
    const int koff = (lane < 16) ? 0 : 2;
    const int nvt  = (n + 15) >> 4;
    float* lat = latent + (size_t)kb * (HH * DHH) + h * DHH;

    for (int t = wave; t < nvt * 4; t += nw) {
        const int vt = t >> 2;
        const int ot = t & 3;
        const int j  = vt * 16 + rc;     // A row this lane supplies
        const bool jv = (j < n);
        const float fj = jv ? f1p[j] : 0.0f;
        const float mj = jv ? mrow[j] : 0.0f;
        const float rj = jv ? rden[j] : 0.0f;
        const int o = ot * 16 + rc;

        v8f acc = {};
        for (int ku = 0; ku < n; ku += 4) {
            const int u0 = ku + koff;
            const int u1 = u0 + 1;
            v2f a, b;
            a.x = (jv && u0 < n) ? expf(lrelu(fj + f2p[u0]) - mj) * rj : 0.0f;
            a.y = (jv && u1 < n) ? expf(lrelu(fj + f2p[u1]) - mj) * rj : 0.0f;
            b.x = (u0 < n) ? EW[((size_t)h * VV + sidx[u0]) * DHH + o] : 0.0f;
            b.y = (u1 < n) ? EW[((size_t)h * VV + sidx[u1]) * DHH + o] : 0.0f;
            acc = wmma_f32_16x16x4(a, b, acc);
        }
        for (int r = 0; r < 8; ++r) {
            int row = vt * 16 + r + ((lane < 16) ? 0 : 8);
            if (row < n) {
                float val = acc[r];
                val = (val > 0.0f) ? val : (expf(val) - 1.0f);  // elu
                atomicMax(lat + ot * 16 + rc, val);
            }
        }
    }
}

// ---------------------------------------------------------------------------
// Kernel 5: final latent attention pool + dense + elu. Tiny: one block per b.
// ---------------------------------------------------------------------------
__global__ void malgat_final_kernel(const float* __restrict__ cls_w,
                                    const float* __restrict__ ac1,
                                    const float* __restrict__ ac2,
                                    const float* __restrict__ dW,
                                    const float* __restrict__ db,
                                    const float* __restrict__ latent,
                                    float* __restrict__ out) {
    __shared__ float hr[3][HH * DHH];
    __shared__ float pooled[HH * DHH];
    __shared__ float attnw[3];
    const int b = blockIdx.x;
    const int tid = threadIdx.x;
    const int F = HH * DHH;  // 256

    for (int f = tid; f < F; f += blockDim.x) {
        hr[0][f] = cls_w[f];
        hr[1][f] = latent[(size_t)(0 * BB + b) * F + f];
        hr[2][f] = latent[(size_t)(1 * BB + b) * F + f];
    }
    __syncthreads();

    if (tid == 0) {
        float dc = 0.0f;
        for (int f = 0; f < F; ++f) dc += hr[0][f] * ac1[f];
        float e[3], mx = FNEG;
        for (int k = 0; k < 3; ++k) {
            float s = 0.0f;
            for (int f = 0; f < F; ++f) s += hr[k][f] * ac2[f];
            e[k] = lrelu(dc + s);
            mx = fmaxf(mx, e[k]);
        }
        float z = 0.0f;
        for (int k = 0; k < 3; ++k) { e[k] = expf(e[k] - mx); z += e[k]; }
        for (int k = 0; k < 3; ++k) attnw[k] = e[k] / z;
    }
    __syncthreads();

    for (int f = tid; f < F; f += blockDim.x)
        pooled[f] = attnw[0] * hr[0][f] + attnw[1] * hr[1][f] + attnw[2] * hr[2][f];
    __syncthreads();

    if (tid < PP) {
        float acc = db[tid];
        for (int f = 0; f < F; ++f) acc += pooled[f] * dW[f * PP + tid];
        out[b * PP + tid] = (acc > 0.0f) ? acc : (expf(acc) - 1.0f);
    }
}

// ---------------------------------------------------------------------------
extern "C" void kernel_launch(void* const* d_in, const int* in_sizes, int n_in,
                              void* d_out, int out_size, void* d_ws, size_t ws_size,
                              hipStream_t stream) {
    (void)in_sizes; (void)n_in; (void)out_size; (void)ws_size;

    const float* x       = (const float*)d_in[0];  // (K,B,V)
    const float* E       = (const float*)d_in[1];  // (V,D)
    const float* W       = (const float*)d_in[2];  // (H,D,DH)
    const float* a1      = (const float*)d_in[3];  // (H,DH)
    const float* a2      = (const float*)d_in[4];  // (H,DH)
    const float* cls_w   = (const float*)d_in[5];  // (H*DH,)
    const float* ac1     = (const float*)d_in[6];  // (H*DH,)
    const float* ac2     = (const float*)d_in[7];  // (H*DH,)
    const float* dense_W = (const float*)d_in[8];  // (H*DH,P)
    const float* dense_b = (const float*)d_in[9];  // (P,)
    float* out = (float*)d_out;                    // (B,P)

    // Workspace carve-up (all fp32-aligned)
    float* ws      = (float*)d_ws;
    float* EW      = ws;                       // H*V*DH = 512000
    float* f1      = EW + (size_t)HH * VV * DHH;      // 8000
    float* f2      = f1 + (size_t)HH * VV;            // 8000
    float* latent  = f2 + (size_t)HH * VV;            // K*B*256 = 2048
    int*   idx     = (int*)(latent + (size_t)KK * BB * HH * DHH);  // K*B*V ints
    int*   counts  = idx + (size_t)KK * BB * VV;                   // K*B ints

    // 1) EW = E @ W[h]  (fp32 WMMA)
    malgat_ew_kernel<<<dim3(VV / 16, HH), 128, 0, stream>>>(E, W, EW);

    // 2) f1/f2 projections
    malgat_f12_kernel<<<(HH * VV + 255) / 256, 256, 0, stream>>>(EW, a1, a2, f1, f2);

    // 3) deterministic presence compaction + latent-max init
    malgat_compact_kernel<<<KK * BB, 64, 0, stream>>>(x, idx, counts, latent);

    // 4) compacted masked attention per (k,b,h) (fp32 WMMA for attn @ EW)
    malgat_attn_kernel<<<dim3(KK * BB, HH), 256, 0, stream>>>(EW, f1, f2, idx,
                                                              counts, latent);

    // 5) final pooling + dense head
    malgat_final_kernel<<<BB, 128, 0, stream>>>(cls_w, ac1, ac2, dense_W, dense_b,
                                                latent, out);
}